// HCTDAFormer_76828374991203
// MI455X (gfx1250) — compile-verified
//
#include <hip/hip_runtime.h>
#include <math.h>

// ---------------------------------------------------------------------------
// HCTDAFormer block for MI455X (gfx1250, wave32). All GEMMs run through
// V_WMMA_F32_16X16X4_F32 (fp32 matrix pipe) to match fp32 reference numerics.
// GEMM: 32x64 tile per wave (8 WMMA per K-step, A-pairs loaded as b64),
// 4 waves/block -> 128x64 per block.
// Workspace: 3 x 100.7MB big buffers + keysmT + kv + small vectors (~306 MB).
// ---------------------------------------------------------------------------

typedef __attribute__((ext_vector_type(2))) float v2f;
typedef __attribute__((ext_vector_type(8))) float v8f;

#define DB   16
#define DC   128
#define DN   1024
#define DL   12
#define DNL  (DN * DL)      // 12288 spatial points per batch (s = n*L + l)
#define DPB  (DC * DNL)     // 1572864 elems per batch in a big tensor
#define DTOT ((long)DB * DPB) // 25165824
#define DCR  8              // C / r
#define DAPT 10

// ------------------------------ reductions ---------------------------------
__device__ __forceinline__ float blockReduceSum(float v, float* sh, int n) {
  int tid = threadIdx.x;
  sh[tid] = v; __syncthreads();
  for (int s = n >> 1; s > 0; s >>= 1) {
    if (tid < s) sh[tid] += sh[tid + s];
    __syncthreads();
  }
  float r = sh[0]; __syncthreads();
  return r;
}
__device__ __forceinline__ float blockReduceMax(float v, float* sh, int n) {
  int tid = threadIdx.x;
  sh[tid] = v; __syncthreads();
  for (int s = n >> 1; s > 0; s >>= 1) {
    if (tid < s) sh[tid] = fmaxf(sh[tid], sh[tid + s]);
    __syncthreads();
  }
  float r = sh[0]; __syncthreads();
  return r;
}

// --------------------- A_p = softmax(relu(nv1 @ nv2)) ----------------------
__global__ __launch_bounds__(256) void ap_kernel(
    const float* __restrict__ nv1, const float* __restrict__ nv2,
    float* __restrict__ Ap, float* __restrict__ rs) {
  __shared__ float sh[256];
  __shared__ float v1[DAPT];
  int row = blockIdx.x, tid = threadIdx.x;
  if (tid < DAPT) v1[tid] = nv1[row * DAPT + tid];
  __syncthreads();
  float vals[4];
  float mx = -3.4e38f;
#pragma unroll
  for (int j = 0; j < 4; j++) {
    int col = tid + j * 256;
    float a = 0.f;
#pragma unroll
    for (int i = 0; i < DAPT; i++) a += v1[i] * nv2[i * DN + col];
    a = fmaxf(a, 0.f);
    vals[j] = a;
    mx = fmaxf(mx, a);
  }
  mx = blockReduceMax(mx, sh, 256);
  float se = 0.f;
#pragma unroll
  for (int j = 0; j < 4; j++) { vals[j] = expf(vals[j] - mx); se += vals[j]; }
  se = blockReduceSum(se, sh, 256);
  float inv = 1.f / se;
  float rsum = 0.f;
#pragma unroll
  for (int j = 0; j < 4; j++) {
    float p = vals[j] * inv;
    Ap[(long)row * DN + tid + j * 256] = p;
    rsum += p;
  }
  rsum = blockReduceSum(rsum, sh, 256);
  if (tid == 0) rs[row] = rsum;
}

// ---- key_sm = softmax(memory/scale, last dim), stored transposed [L][C][N] --
__global__ __launch_bounds__(128) void keysm_kernel(
    const float* __restrict__ mem, float* __restrict__ kT) {
  __shared__ float sh[128];
  int rid = blockIdx.x;           // l*N + n
  int l = rid / DN, n = rid % DN;
  int c = threadIdx.x;
  const float inv_scale = 1.f / sqrtf((float)DC);
  float v = mem[(long)rid * DC + c] * inv_scale;
  float mx = blockReduceMax(v, sh, 128);
  float e = expf(v - mx);
  float s = blockReduceSum(e, sh, 128);
  kT[((long)l * DC + c) * DN + n] = e / s;
}

// --------------------------- SE pooling / FC -------------------------------
__global__ __launch_bounds__(256) void pool_kernel(
    const float* __restrict__ in, float* __restrict__ out,
    long bstride, long cstride, long estride) {
  __shared__ float sh[256];
  int c = blockIdx.x, b = blockIdx.y, tid = threadIdx.x;
  const float* p = in + (long)b * bstride + (long)c * cstride;
  float s = 0.f;
  for (int i = tid; i < DNL; i += 256) s += p[(long)i * estride];
  s = blockReduceSum(s, sh, 256);
  if (tid == 0) out[b * DC + c] = s * (1.f / (float)DNL);
}

__global__ __launch_bounds__(128) void se_fc_kernel(
    const float* __restrict__ W1, const float* __restrict__ W2,
    const float* __restrict__ s, float* __restrict__ s2) {
  __shared__ float sv[DC];
  __shared__ float t[DCR];
  int b = blockIdx.x, tid = threadIdx.x;
  sv[tid] = s[b * DC + tid];
  __syncthreads();
  if (tid < DCR) {
    float a = 0.f;
    for (int i = 0; i < DC; i++) a += W1[tid * DC + i] * sv[i];
    t[tid] = fmaxf(a, 0.f);
  }
  __syncthreads();
  float a = 0.f;
#pragma unroll
  for (int j = 0; j < DCR; j++) a += W2[tid * DCR + j] * t[j];
  s2[b * DC + tid] = 1.f / (1.f + expf(-a));
}

// -------- q: SE scale then softmax over contiguous C (layout [B][NL][C]) ----
__global__ __launch_bounds__(128) void qsm_kernel(
    float* __restrict__ q, const float* __restrict__ s2) {
  __shared__ float sh[128];
  long row = blockIdx.x;  // b*DNL + s
  int b = (int)(row / DNL);
  int c = threadIdx.x;
  const float inv_scale = 1.f / sqrtf((float)DC);
  float v = q[row * DC + c] * s2[b * DC + c] * inv_scale;
  float mx = blockReduceMax(v, sh, 128);
  float e = expf(v - mx);
  float s = blockReduceSum(e, sh, 128);
  q[row * DC + c] = e / s;
}

// v: SE scale elementwise (layout [B][NL][C])
__global__ void vscale_kernel(float* __restrict__ v, const float* __restrict__ s2) {
  for (long i = (long)blockIdx.x * blockDim.x + threadIdx.x; i < DTOT;
       i += (long)gridDim.x * blockDim.x) {
    int b = (int)(i / DPB);
    int c = (int)(i % DC);
    v[i] *= s2[b * DC + c];
  }
}

// h path: SE scale + affine residual (h*w + b + h) + skip x  (natural layout)
__global__ void coscale_kernel(const float* __restrict__ h2,
                               const float* __restrict__ s2,
                               const float* __restrict__ wt,
                               const float* __restrict__ bt,
                               const float* __restrict__ x,
                               float* __restrict__ y0) {
  for (long i = (long)blockIdx.x * blockDim.x + threadIdx.x; i < DTOT;
       i += (long)gridDim.x * blockDim.x) {
    int b = (int)(i / DPB);
    long rem = i % DPB;
    int c = (int)(rem / DNL);
    float hv = h2[i] * s2[b * DC + c];
    float t = hv * wt[rem] + bt[rem] + hv;
    y0[i] = t + x[i];
  }
}

// ---------------------- LayerNorm over (C,N) per (b,l) ---------------------
__global__ __launch_bounds__(256) void ln_stats_kernel(
    const float* __restrict__ in, float* __restrict__ stats) {
  __shared__ float sh[256];
  int g = blockIdx.x;
  int b = g / DL, l = g % DL;
  const float* p = in + (long)b * DPB + l;
  float s = 0.f, ss = 0.f;
  for (int u = threadIdx.x; u < DC * DN; u += 256) {
    float v = p[(long)u * DL];
    s += v; ss += v * v;
  }
  s = blockReduceSum(s, sh, 256);
  ss = blockReduceSum(ss, sh, 256);
  if (threadIdx.x == 0) {
    float m = s / (float)(DC * DN);
    stats[g * 2] = m;
    stats[g * 2 + 1] = ss / (float)(DC * DN) - m * m;
  }
}

__global__ void ln_apply_kernel(const float* __restrict__ in,
                                float* __restrict__ out,
                                const float* __restrict__ stats) {
  for (long i = (long)blockIdx.x * blockDim.x + threadIdx.x; i < DTOT;
       i += (long)gridDim.x * blockDim.x) {
    int b = (int)(i / DPB);
    int l = (int)(i % DL);
    int g = b * DL + l;
    float m = stats[g * 2], v = stats[g * 2 + 1];
    out[i] = (in[i] - m) * rsqrtf(v + 1e-5f);
  }
}

// --------------------------- generic WMMA GEMM -----------------------------
// D[row,col] = sum_k A[row,k] * B[k,col]  (per z-slice), fp32 WMMA 16x16x4.
// addr = z1*z?1 + z2*z?2 + row*rs + k1*k1s + k2 (+ col*cs). A k2-stride is
// REQUIRED to be 1 with ars/ak1 even (contiguous, 8B-aligned pair -> b64 load).
// K = Ktot = K1*K2, K2 % 4 == 0. Wave tile: 32(M) x 64(N); block: 128 x 64.
struct GemmEpi {
  int mode;               // 0 none,1 bias+relu,2 bias,3 sigmoid(.)*aux,
                          // 4 A_h/A_f dual store, 5 +rsvec[row]*aux, 6 final GLU
  const float* bias;      // per-row (output channel)
  const float* aux;       // C-addressed tensor
  float* out2;            // C-addressed second output (A_f)
  const float* apmat;     // row*apld + col (A_p)
  const float* rsvec;     // per global row
  const float* wt;        // row*wrs + col (weight tensor)
  const float* bt;        // row*wrs + col (bias tensor)
  long apld;
  long wrs;
  float kscale;
};

__global__ __launch_bounds__(128) void gemm_wmma_kernel(
    const float* __restrict__ A, const float* __restrict__ B,
    float* __restrict__ C, int Ktot, int K2, int Z2,
    long ars, long ak1, long az1, long az2,
    long bcs, long bk1, long bk2, long bz1, long bz2,
    long crs, long ccs, long cz1, long cz2, GemmEpi e) {
  const int lane = threadIdx.x & 31;
  const int wid = threadIdx.x >> 5;
  const int m0 = (blockIdx.y * 4 + wid) << 5;   // 32 rows per wave
  const int n0 = blockIdx.x << 6;               // 64 cols per wave
  const int z = blockIdx.z;
  const int z1 = z / Z2, z2 = z % Z2;
  const float* Ab = A + (long)z1 * az1 + (long)z2 * az2;
  const float* Bb = B + (long)z1 * bz1 + (long)z2 * bz2;
  const long cbase = (long)z1 * cz1 + (long)z2 * cz2;

  const int halfsel = lane >> 4;  // 0|1 : lanes 16-31 hold K+2 / M+8 halves
  const int kp = halfsel << 1;
  const int nl = lane & 15;

  v8f acc[2][4];
#pragma unroll
  for (int h = 0; h < 2; h++)
#pragma unroll
    for (int j = 0; j < 4; j++)
      acc[h][j] = (v8f){0.f, 0.f, 0.f, 0.f, 0.f, 0.f, 0.f, 0.f};

  long ao0 = (long)(m0 + nl) * ars + kp;   // A k2-stride == 1
  long ao1 = ao0 + 16 * ars;
  long bko = (long)kp * bk2;
  const long aFix = ak1 - (long)K2;
  const long bFix = bk1 - (long)K2 * bk2;
  int k2 = 0;
  for (int kk = 0; kk < Ktot; kk += 4) {
    v2f af0 = *(const v2f*)(Ab + ao0);     // b64: two contiguous A elems
    v2f af1 = *(const v2f*)(Ab + ao1);
#pragma unroll
    for (int j = 0; j < 4; j++) {
      long bo = bko + (long)(n0 + (j << 4) + nl) * bcs;
      v2f bf;
      bf.x = Bb[bo];
      bf.y = Bb[bo + bk2];
      acc[0][j] = __builtin_amdgcn_wmma_f32_16x16x4_f32(
          false, af0, false, bf, (short)0, acc[0][j], false, false);
      acc[1][j] = __builtin_amdgcn_wmma_f32_16x16x4_f32(
          false, af1, false, bf, (short)0, acc[1][j], false, false);
    }
    ao0 += 4; ao1 += 4; bko += 4 * bk2;
    k2 += 4;
    if (k2 == K2) { k2 = 0; ao0 += aFix; ao1 += aFix; bko += bFix; }
  }

#pragma unroll
  for (int h = 0; h < 2; h++) {
#pragma unroll
    for (int j = 0; j < 4; j++) {
      const int col = n0 + (j << 4) + nl;
#pragma unroll
      for (int i = 0; i < 8; i++) {
        const int row = m0 + (h << 4) + i + (halfsel << 3);
        const long caddr = cbase + (long)row * crs + (long)col * ccs;
        float v = acc[h][j][i] * e.kscale;
        switch (e.mode) {
          case 0: C[caddr] = v; break;
          case 1: C[caddr] = fmaxf(v + e.bias[row], 0.f); break;
          case 2: C[caddr] = v + e.bias[row]; break;
          case 3: {
            float g = 1.f / (1.f + expf(-(v + e.bias[row])));
            C[caddr] = e.aux[caddr] * g;
          } break;
          case 4: {
            C[caddr] = v;
            e.out2[caddr] = v + e.apmat[(long)row * e.apld + col];
          } break;
          case 5: C[caddr] = v + e.rsvec[row] * e.aux[caddr]; break;
          case 6: {
            float t = v + e.bias[row] + e.aux[caddr];
            long w = (long)row * e.wrs + col;
            C[caddr] = t * e.wt[w] + e.bt[w] + t;
          } break;
        }
      }
    }
  }
}

// ---------------------------------------------------------------------------
extern "C" void kernel_launch(void* const* d_in, const int* in_sizes, int n_in,
                              void* d_out, int out_size, void* d_ws,
                              size_t ws_size, hipStream_t stream) {
  (void)in_sizes; (void)n_in; (void)out_size; (void)ws_size;
  const float* x    = (const float*)d_in[0];
  const float* Wq   = (const float*)d_in[1];
  const float* bq   = (const float*)d_in[2];
  const float* Wq1  = (const float*)d_in[3];
  const float* Wq2  = (const float*)d_in[4];
  const float* Wv   = (const float*)d_in[5];
  const float* bv   = (const float*)d_in[6];
  const float* Wv1  = (const float*)d_in[7];
  const float* Wv2  = (const float*)d_in[8];
  const float* Wco  = (const float*)d_in[9];
  const float* bco  = (const float*)d_in[10];
  const float* Wc1  = (const float*)d_in[11];
  const float* Wc2  = (const float*)d_in[12];
  const float* mem  = (const float*)d_in[13];
  const float* nv1  = (const float*)d_in[14];
  const float* nv2  = (const float*)d_in[15];
  const float* wt   = (const float*)d_in[16];
  const float* bt   = (const float*)d_in[17];
  const float* Wg1  = (const float*)d_in[18];
  const float* bg1  = (const float*)d_in[19];
  const float* Wg2  = (const float*)d_in[20];
  const float* bg2  = (const float*)d_in[21];
  const float* Wg3  = (const float*)d_in[22];
  const float* bg3  = (const float*)d_in[23];

  float* out  = (float*)d_out;
  float* yOut = out;
  float* ApO  = out + DTOT;
  float* AhO  = ApO + (long)DN * DN;
  float* AfO  = AhO + (long)DB * DN * DN;

  float* ws    = (float*)d_ws;
  float* qbuf  = ws;                              // [B][NL][C] then natural
  float* vbuf  = qbuf + DTOT;                     // [B][NL][C] then natural
  float* hbuf  = vbuf + DTOT;                     // [B][NL][C] then natural
  float* kT    = hbuf + DTOT;                     // [L][C][N]
  float* kvb   = kT + (long)DL * DC * DN;         // [B*L][C][C]
  float* rsv   = kvb + (long)DB * DL * DC * DC;   // [N]
  float* sbuf  = rsv + DN;                        // [B*C]
  float* s2buf = sbuf + DB * DC;                  // [B*C]
  float* stats = s2buf + DB * DC;                 // [B*L][2]

  const long LC = (long)DL * DC;        // 1536
  const long CN = (long)DC * DN;        // 131072
  const long KVZ = (long)DC * DC;       // 16384
  GemmEpi ep;

  // A_p + rowsums ; key softmax (transposed store)
  ap_kernel<<<DN, 256, 0, stream>>>(nv1, nv2, ApO, rsv);
  keysm_kernel<<<DL * DN, 128, 0, stream>>>(mem, kT);

  // q = relu(Wq x + bq)  -> qbuf [B][NL][C]
  ep = GemmEpi{1, bq, nullptr, nullptr, nullptr, nullptr, nullptr, nullptr, 0, 0, 1.f};
  gemm_wmma_kernel<<<dim3(DNL / 64, 1, DB), 128, 0, stream>>>(
      Wq, x, qbuf, DC, DC, 1,
      DC, 0, 0, 0,             // A = Wq[o][c]
      1, 0, DNL, DPB, 0,       // B = x[c][s] natural
      1, DC, DPB, 0, ep);      // C = qbuf: s*C + o
  pool_kernel<<<dim3(DC, DB), 256, 0, stream>>>(qbuf, sbuf, DPB, 1, DC);
  se_fc_kernel<<<DB, 128, 0, stream>>>(Wq1, Wq2, sbuf, s2buf);
  qsm_kernel<<<DB * DNL, 128, 0, stream>>>(qbuf, s2buf);

  // v = relu(Wv x + bv) -> vbuf ; SE scale
  ep = GemmEpi{1, bv, nullptr, nullptr, nullptr, nullptr, nullptr, nullptr, 0, 0, 1.f};
  gemm_wmma_kernel<<<dim3(DNL / 64, 1, DB), 128, 0, stream>>>(
      Wv, x, vbuf, DC, DC, 1,
      DC, 0, 0, 0,
      1, 0, DNL, DPB, 0,
      1, DC, DPB, 0, ep);
  pool_kernel<<<dim3(DC, DB), 256, 0, stream>>>(vbuf, sbuf, DPB, 1, DC);
  se_fc_kernel<<<DB, 128, 0, stream>>>(Wv1, Wv2, sbuf, s2buf);
  vscale_kernel<<<4096, 256, 0, stream>>>(vbuf, s2buf);

  // A_h[b,n,m] = sum_{l,c} q_sm * key_sm ; A_f = A_h + A_p
  ep = GemmEpi{4, nullptr, nullptr, AfO, ApO, nullptr, nullptr, nullptr, DN, 0, 1.f};
  gemm_wmma_kernel<<<dim3(DN / 64, DN / 128, DB), 128, 0, stream>>>(
      qbuf, kT, AhO, DL * DC, DC, 1,
      LC, DC, DPB, 0,          // A = q_sm[b][n][(l,c)]
      1, CN, DN, 0, 0,         // B = kT[(l,c)][m]
      DN, 1, (long)DN * DN, 0, ep);

  // kv[b,l,x,y] = sum_n key_sm[l,n,x] v[b,l,n,y]
  ep = GemmEpi{0, nullptr, nullptr, nullptr, nullptr, nullptr, nullptr, nullptr, 0, 0, 1.f};
  gemm_wmma_kernel<<<dim3(DC / 64, 1, DB * DL), 128, 0, stream>>>(
      kT, vbuf, kvb, DN, DN, DL,
      DN, 0, 0, CN,            // A = kT[l][x][n]
      1, 0, LC, DPB, DC,       // B = v[b][n][l][y]
      DC, 1, DL * KVZ, KVZ, ep);

  // h = q_sm @ kv + rowsum(A_p)[n] * v  -> hbuf [B][NL][C]
  ep = GemmEpi{5, nullptr, vbuf, nullptr, nullptr, rsv, nullptr, nullptr, 0, 0, 1.f};
  gemm_wmma_kernel<<<dim3(DC / 64, DN / 128, DB * DL), 128, 0, stream>>>(
      qbuf, kvb, hbuf, DC, DC, DL,
      LC, 0, DPB, DC,          // A = q_sm[b][n][l][x]
      1, 0, DC, DL * KVZ, KVZ, // B = kv[z][x][y]
      LC, 1, DPB, DC, ep);     // C = h[b][n][l][y]

  // h2 = relu(Wco h + bco) -> qbuf (natural [B,C,N,L]); qbuf is free now
  ep = GemmEpi{1, bco, nullptr, nullptr, nullptr, nullptr, nullptr, nullptr, 0, 0, 1.f};
  gemm_wmma_kernel<<<dim3(DNL / 64, 1, DB), 128, 0, stream>>>(
      Wco, hbuf, qbuf, DC, DC, 1,
      DC, 0, 0, 0,
      DC, 0, 1, DPB, 0,        // B = h[c][s] : s*C + c
      DNL, 1, DPB, 0, ep);     // C natural: o*NL + s
  pool_kernel<<<dim3(DC, DB), 256, 0, stream>>>(qbuf, sbuf, DPB, DNL, 1);
  se_fc_kernel<<<DB, 128, 0, stream>>>(Wc1, Wc2, sbuf, s2buf);
  // SE scale + (h*w + b + h) + x  -> qbuf (in place)
  coscale_kernel<<<4096, 256, 0, stream>>>(qbuf, s2buf, wt, bt, x, qbuf);

  // LayerNorm over (C,N) per (b,l), in place
  ln_stats_kernel<<<DB * DL, 256, 0, stream>>>(qbuf, stats);
  ln_apply_kernel<<<4096, 256, 0, stream>>>(qbuf, qbuf, stats);

  // GLU: t1 = Wg1 y + bg1 -> vbuf ; g = t1 * sigmoid(Wg2 y + bg2) -> vbuf
  ep = GemmEpi{2, bg1, nullptr, nullptr, nullptr, nullptr, nullptr, nullptr, 0, 0, 1.f};
  gemm_wmma_kernel<<<dim3(DNL / 64, 1, DB), 128, 0, stream>>>(
      Wg1, qbuf, vbuf, DC, DC, 1,
      DC, 0, 0, 0,
      1, 0, DNL, DPB, 0,
      DNL, 1, DPB, 0, ep);
  ep = GemmEpi{3, bg2, vbuf, nullptr, nullptr, nullptr, nullptr, nullptr, 0, 0, 1.f};
  gemm_wmma_kernel<<<dim3(DNL / 64, 1, DB), 128, 0, stream>>>(
      Wg2, qbuf, vbuf, DC, DC, 1,
      DC, 0, 0, 0,
      1, 0, DNL, DPB, 0,
      DNL, 1, DPB, 0, ep);
  // g3 = Wg3 g + bg3 ; t = g3 + y ; res = t*w + b + t  -> hbuf
  ep = GemmEpi{6, bg3, qbuf, nullptr, nullptr, nullptr, wt, bt, 0, DNL, 1.f};
  gemm_wmma_kernel<<<dim3(DNL / 64, 1, DB), 128, 0, stream>>>(
      Wg3, vbuf, hbuf, DC, DC, 1,
      DC, 0, 0, 0,
      1, 0, DNL, DPB, 0,
      DNL, 1, DPB, 0, ep);

  // final LayerNorm -> y output
  ln_stats_kernel<<<DB * DL, 256, 0, stream>>>(hbuf, stats);
  ln_apply_kernel<<<4096, 256, 0, stream>>>(hbuf, yOut, stats);
}